// RNNTLoss_49323404427390
// MI455X (gfx1250) — compile-verified
//
#include <hip/hip_runtime.h>
#include <math.h>

typedef __attribute__((ext_vector_type(2))) float v2f;
typedef __attribute__((ext_vector_type(8))) float v8f;

#define NEGV (-1e30f)

// Problem dims (fixed by the reference harness)
#define B_   8
#define T_   128
#define U_   64
#define U1_  65
#define UP_  80    // U+1 padded to a multiple of 16 for WMMA tiles
#define V_   512

// ---------------------------------------------------------------------------
// Kernel 1: per-row max and exp(x - max). Rows beyond rowsValid are zero-filled
// (pad rows of the pred matrix) so the GEMM can run over the full padded tile.
// ---------------------------------------------------------------------------
__global__ __launch_bounds__(128)
void rowexp_kernel(const float* __restrict__ in, float* __restrict__ outE,
                   float* __restrict__ outM, int rowsValid, int rowsTotal)
{
    __shared__ float red[128];
    const int g   = blockIdx.x;            // [0, B*rowsTotal)
    const int b   = g / rowsTotal;
    const int r   = g % rowsTotal;
    const int tid = threadIdx.x;
    float* e = outE + (size_t)g * V_;

    if (r >= rowsValid) {                  // zero pad rows
        for (int v = tid; v < V_; v += 128) e[v] = 0.0f;
        if (tid == 0) outM[g] = 0.0f;
        return;
    }

    const float* x = in + ((size_t)b * rowsValid + r) * V_;
    float m = -INFINITY;
    for (int v = tid; v < V_; v += 128) m = fmaxf(m, x[v]);
    red[tid] = m;
    __syncthreads();
    for (int s = 64; s > 0; s >>= 1) {
        if (tid < s) red[tid] = fmaxf(red[tid], red[tid + s]);
        __syncthreads();
    }
    m = red[0];
    for (int v = tid; v < V_; v += 128) e[v] = expf(x[v] - m);
    if (tid == 0) outM[g] = m;
}

// ---------------------------------------------------------------------------
// Kernel 2: per-wave 16x80 strip of S[b,t,u] = sum_v ET[t,v]*EP[u,v] via
// V_WMMA_F32_16X16X4_F32 (K-loop of 128 steps, A fragment reused over 5 tiles),
// then lse = mt + mu + log(S); emit blank_lp and (masked) lab_lp.
//
// f32 16x4 A fragment (ISA 7.12.2): lane<16 holds {K=0,K=1} of row M=lane;
// lane>=16 holds {K=2,K=3} of row M=lane-16.  B (4x16) mirrors with N=lane&15.
// C/D: vgpr i -> M = i + 8*(lane>=16), N = lane&15.
// ---------------------------------------------------------------------------
__global__ __launch_bounds__(32)
void joint_gemm_kernel(const float* __restrict__ ET, const float* __restrict__ EP,
                       const float* __restrict__ MT, const float* __restrict__ MU,
                       const float* __restrict__ trans, const float* __restrict__ pred,
                       const int* __restrict__ labels, const int* __restrict__ label_lens,
                       float* __restrict__ blank_lp, float* __restrict__ lab_lp)
{
    const int b    = blockIdx.x >> 3;          // batch
    const int t0   = (blockIdx.x & 7) << 4;    // 16-row t tile
    const int lane = threadIdx.x;
    const int half = lane >> 4;                // 0: K pair {0,1}, 1: K pair {2,3}
    const int lr   = lane & 15;
    const int koff = half << 1;

    const float* Abase = ET + ((size_t)(b * T_ + t0 + lr)) * V_;
    const float* Bbase = EP + ((size_t)(b * UP_ + lr)) * V_;   // + j*16*V_ per tile

    v8f acc[5] = {};
    for (int k = 0; k < V_; k += 4) {
        v2f a = *(const v2f*)(Abase + k + koff);
        v2f b0 = *(const v2f*)(Bbase + (size_t)0 * 16 * V_ + k + koff);
        v2f b1 = *(const v2f*)(Bbase + (size_t)1 * 16 * V_ + k + koff);
        v2f b2 = *(const v2f*)(Bbase + (size_t)2 * 16 * V_ + k + koff);
        v2f b3 = *(const v2f*)(Bbase + (size_t)3 * 16 * V_ + k + koff);
        v2f b4 = *(const v2f*)(Bbase + (size_t)4 * 16 * V_ + k + koff);
        acc[0] = __builtin_amdgcn_wmma_f32_16x16x4_f32(false, a, false, b0, (short)0, acc[0], false, false);
        acc[1] = __builtin_amdgcn_wmma_f32_16x16x4_f32(false, a, false, b1, (short)0, acc[1], false, false);
        acc[2] = __builtin_amdgcn_wmma_f32_16x16x4_f32(false, a, false, b2, (short)0, acc[2], false, false);
        acc[3] = __builtin_amdgcn_wmma_f32_16x16x4_f32(false, a, false, b3, (short)0, acc[3], false, false);
        acc[4] = __builtin_amdgcn_wmma_f32_16x16x4_f32(false, a, false, b4, (short)0, acc[4], false, false);
    }

    // Epilogue: per accumulated element compute log-probs and store.
    const int llen = label_lens[b];
    #pragma unroll
    for (int j = 0; j < 5; ++j) {
        const int u = j * 16 + lr;
        if (u >= U1_) continue;
        const float mu   = MU[b * UP_ + u];
        const float pbl  = pred[((size_t)(b * U1_ + u)) * V_ + 0];
        #pragma unroll
        for (int i = 0; i < 8; ++i) {
            const int t = t0 + i + 8 * half;
            const float S   = acc[j][i];
            const float lse = MT[b * T_ + t] + mu + logf(S);
            const float tbl = trans[((size_t)(b * T_ + t)) * V_ + 0];
            blank_lp[((size_t)(b * T_ + t)) * U1_ + u] = tbl + pbl - lse;
            if (u < U_) {
                float lv = NEGV;
                if (u < llen) {
                    const int l = labels[b * U_ + u];
                    lv = trans[((size_t)(b * T_ + t)) * V_ + l]
                       + pred[((size_t)(b * U1_ + u)) * V_ + l] - lse;
                }
                lab_lp[((size_t)(b * T_ + t)) * U_ + u] = lv;
            }
        }
    }
}

// ---------------------------------------------------------------------------
// Kernel 3: anti-diagonal wavefront alpha DP per batch, lattice in LDS.
// alpha[t,u] = logaddexp(alpha[t-1,u]+blank[t-1,u], alpha[t,u-1]+lab[t,u-1])
// ---------------------------------------------------------------------------
__global__ __launch_bounds__(128)
void alpha_kernel(const float* __restrict__ blank_lp, const float* __restrict__ lab_lp,
                  const int* __restrict__ act_lens, const int* __restrict__ label_lens,
                  float* __restrict__ nll)
{
    __shared__ float alpha[T_ * U1_];     // 33,280 B of the 320 KB LDS
    const int b   = blockIdx.x;
    const int tid = threadIdx.x;
    const float* blk = blank_lp + (size_t)b * T_ * U1_;
    const float* lab = lab_lp   + (size_t)b * T_ * U_;

    for (int d = 0; d < T_ + U1_ - 1; ++d) {
        const int tmin = (d - (U1_ - 1)) > 0 ? (d - (U1_ - 1)) : 0;
        const int tmax = d < (T_ - 1) ? d : (T_ - 1);
        const int t = tmin + tid;
        if (t <= tmax) {
            const int u = d - t;
            float val;
            if (t == 0 && u == 0) {
                val = 0.0f;
            } else if (t == 0) {
                val = alpha[u - 1] + lab[u - 1];
            } else if (u == 0) {
                val = alpha[(t - 1) * U1_] + blk[(t - 1) * U1_];
            } else {
                const float a1 = alpha[(t - 1) * U1_ + u] + blk[(t - 1) * U1_ + u];
                const float a2 = alpha[t * U1_ + (u - 1)] + lab[t * U_ + (u - 1)];
                const float m  = fmaxf(a1, a2);
                val = m + log1pf(expf(fminf(a1, a2) - m));
            }
            alpha[t * U1_ + u] = val;
        }
        __syncthreads();
    }

    if (tid == 0) {
        const int tl = act_lens[b] - 1;
        const int ul = label_lens[b];
        nll[b] = -(alpha[tl * U1_ + ul] + blk[tl * U1_ + ul]);
    }
}

// ---------------------------------------------------------------------------
// Kernel 4: final batch-sum into d_out[0] (d_out is poisoned by the harness).
// ---------------------------------------------------------------------------
__global__ __launch_bounds__(32)
void reduce_kernel(const float* __restrict__ nll, float* __restrict__ out)
{
    if (threadIdx.x == 0) {
        float s = 0.0f;
        #pragma unroll
        for (int i = 0; i < B_; ++i) s += nll[i];
        out[0] = s;
    }
}

// ---------------------------------------------------------------------------
extern "C" void kernel_launch(void* const* d_in, const int* in_sizes, int n_in,
                              void* d_out, int out_size, void* d_ws, size_t ws_size,
                              hipStream_t stream)
{
    (void)in_sizes; (void)n_in; (void)out_size; (void)ws_size;

    const float* trans      = (const float*)d_in[0];   // (B, T,  V)
    const float* pred       = (const float*)d_in[1];   // (B, U+1, V)
    const int*   labels     = (const int*)d_in[2];     // (B, U)
    const int*   act_lens   = (const int*)d_in[3];     // (B,)
    const int*   label_lens = (const int*)d_in[4];     // (B,)
    float*       out        = (float*)d_out;           // (1,)

    // Workspace layout (floats)
    float* ws = (float*)d_ws;
    float* ET    = ws;                         // B*T *V  = 524288
    float* EP    = ET    + (size_t)B_ * T_  * V_;   // B*UP*V  = 327680
    float* MT    = EP    + (size_t)B_ * UP_ * V_;   // B*T     = 1024
    float* MU    = MT    + (size_t)B_ * T_;         // B*UP    = 640
    float* blank = MU    + (size_t)B_ * UP_;        // B*T*U1  = 66560
    float* labp  = blank + (size_t)B_ * T_ * U1_;   // B*T*U   = 65536
    float* nll   = labp  + (size_t)B_ * T_ * U_;    // B       = 8

    // 1) row max + exp for trans (T rows/batch) and pred (padded to UP rows)
    rowexp_kernel<<<B_ * T_,  128, 0, stream>>>(trans, ET, MT, T_,  T_);
    rowexp_kernel<<<B_ * UP_, 128, 0, stream>>>(pred,  EP, MU, U1_, UP_);

    // 2) fp32 WMMA GEMM -> logsumexp -> blank/label log-probs
    joint_gemm_kernel<<<B_ * (T_ / 16), 32, 0, stream>>>(
        ET, EP, MT, MU, trans, pred, labels, label_lens, blank, labp);

    // 3) alpha lattice wavefront per batch element
    alpha_kernel<<<B_, 128, 0, stream>>>(blank, labp, act_lens, label_lens, nll);

    // 4) batch sum
    reduce_kernel<<<1, 32, 0, stream>>>(nll, out);
}